// GaussianRayTracer_27453430956363
// MI455X (gfx1250) — compile-verified
//
#include <hip/hip_runtime.h>
#include <hip/hip_bf16.h>
#include <math.h>

// ---------------------------------------------------------------------------
// Fused 3D-Gaussian ray tracer for gfx1250 (MI455X, wave32, WMMA).
//   grid  : 1024 blocks (16 rays per block)
//   block : 512 threads = 16 wave32 waves
//   LDS   : dynamic ~219 KB (Bmat 12x1040 + gf 1024x9 + interleaved t/la)
// Pair evaluation (dot_od, dd) on V_WMMA_F32_16X16X4_F32 (12 WMMAs / wave,
// dd chained through the C accumulator). Compositing uses one u64-key
// compare per pair (stable front-to-back order without sorting), with all
// transcendentals in base-2 (raw v_log_f32 / v_exp_f32, no branchy libm).
// ---------------------------------------------------------------------------

typedef __attribute__((ext_vector_type(2))) float v2f;
typedef __attribute__((ext_vector_type(8))) float v8f;

#define NG    1024          // gaussians
#define RPB   16            // rays per block
#define TS2   2060          // words per ray row of interleaved (t,la) pairs
#define BSTR  1040          // padded stride for B-matrix rows
#define GSTR  9             // per-gaussian feature stride (conflict-free mod 64)

#define SM_BM   (12 * BSTR)            // 12480 floats
#define SM_GF   (NG * GSTR)            //  9216 floats
#define SM_TL   (RPB * TS2)            // 32960 floats (t,la interleaved)
#define SM_TOT  (SM_BM + SM_GF + SM_TL)   // 54656 floats = 218624 B

#define NEG_HALF_LOG2E  (-0.72134752044448170f)  // -0.5*log2(e)
#define LOG2_TMIN       (-9.96578428466208700f)  // log2(1e-3)

__device__ __forceinline__ float wave_sum(float v) {
  #pragma unroll
  for (int m = 16; m > 0; m >>= 1) v += __shfl_xor(v, m, 32);
  return v;
}

__global__ __launch_bounds__(512, 1)
void gs_raytrace_kernel(const float* __restrict__ xyz,
                        const float* __restrict__ rot,
                        const float* __restrict__ sca,
                        const float* __restrict__ opa,
                        const float* __restrict__ rgb,
                        const float* __restrict__ nrm,
                        const float* __restrict__ bg,
                        const float* __restrict__ rorg,
                        const float* __restrict__ rdir,
                        float* __restrict__ out) {
  extern __shared__ float sm[];
  float* Bm = sm;                     // [12][BSTR]  WMMA B-matrix features
  float* gf = sm + SM_BM;             // [NG][GSTR]  rgb(3) nrmU(3) oo opa pad
  float* tl = gf + SM_GF;             // [RPB][TS2]  interleaved (t, la) pairs
  __shared__ float sdir[RPB * 3];

  const int tid  = threadIdx.x;
  const int lane = tid & 31;
  const int wv   = tid >> 5;

  // ---------------- per-gaussian setup (redundant per block; tiny) ---------
  const float rox = rorg[0], roy = rorg[1], roz = rorg[2];
  for (int n = tid; n < NG; n += 512) {
    float qw = rot[n * 4 + 0], qx = rot[n * 4 + 1];
    float qy = rot[n * 4 + 2], qz = rot[n * 4 + 3];
    float qn = rsqrtf(qw * qw + qx * qx + qy * qy + qz * qz + 1e-12f);
    qw *= qn; qx *= qn; qy *= qn; qz *= qn;
    float r00 = 1.f - 2.f * (qy * qy + qz * qz);
    float r01 = 2.f * (qx * qy - qw * qz);
    float r02 = 2.f * (qx * qz + qw * qy);
    float r10 = 2.f * (qx * qy + qw * qz);
    float r11 = 1.f - 2.f * (qx * qx + qz * qz);
    float r12 = 2.f * (qy * qz - qw * qx);
    float r20 = 2.f * (qx * qz - qw * qy);
    float r21 = 2.f * (qy * qz + qw * qx);
    float r22 = 1.f - 2.f * (qx * qx + qy * qy);
    float is0 = 1.f / sca[n * 3 + 0];
    float is1 = 1.f / sca[n * 3 + 1];
    float is2 = 1.f / sca[n * 3 + 2];
    // Minv[i][j] = R[j][i] / s_i
    float m00 = r00 * is0, m01 = r10 * is0, m02 = r20 * is0;
    float m10 = r01 * is1, m11 = r11 * is1, m12 = r21 * is1;
    float m20 = r02 * is2, m21 = r12 * is2, m22 = r22 * is2;
    float ox = rox - xyz[n * 3 + 0];
    float oy = roy - xyz[n * 3 + 1];
    float oz = roz - xyz[n * 3 + 2];
    float l0 = m00 * ox + m01 * oy + m02 * oz;
    float l1 = m10 * ox + m11 * oy + m12 * oz;
    float l2 = m20 * ox + m21 * oy + m22 * oz;
    float v0 = m00 * l0 + m10 * l1 + m20 * l2;   // v = Minv^T o_loc
    float v1 = m01 * l0 + m11 * l1 + m21 * l2;
    float v2 = m02 * l0 + m12 * l1 + m22 * l2;
    float oo = l0 * l0 + l1 * l1 + l2 * l2;
    float Q00 = m00 * m00 + m10 * m10 + m20 * m20;
    float Q11 = m01 * m01 + m11 * m11 + m21 * m21;
    float Q22 = m02 * m02 + m12 * m12 + m22 * m22;
    float Q01 = m00 * m01 + m10 * m11 + m20 * m21;
    float Q02 = m00 * m02 + m10 * m12 + m20 * m22;
    float Q12 = m01 * m02 + m11 * m12 + m21 * m22;
    Bm[0 * BSTR + n] = v0;
    Bm[1 * BSTR + n] = v1;
    Bm[2 * BSTR + n] = v2;
    Bm[3 * BSTR + n] = 0.f;
    Bm[4 * BSTR + n] = Q00;
    Bm[5 * BSTR + n] = Q11;
    Bm[6 * BSTR + n] = Q22;
    Bm[7 * BSTR + n] = 2.f * Q01;
    Bm[8 * BSTR + n] = 2.f * Q02;
    Bm[9 * BSTR + n] = 2.f * Q12;
    Bm[10 * BSTR + n] = 0.f;
    Bm[11 * BSTR + n] = 0.f;
    float nx = nrm[n * 3 + 0], ny = nrm[n * 3 + 1], nz = nrm[n * 3 + 2];
    float nn = rsqrtf(nx * nx + ny * ny + nz * nz + 1e-12f);
    gf[n * GSTR + 0] = rgb[n * 3 + 0];
    gf[n * GSTR + 1] = rgb[n * 3 + 1];
    gf[n * GSTR + 2] = rgb[n * 3 + 2];
    gf[n * GSTR + 3] = nx * nn;
    gf[n * GSTR + 4] = ny * nn;
    gf[n * GSTR + 5] = nz * nn;
    gf[n * GSTR + 6] = oo;
    gf[n * GSTR + 7] = opa[n];
    gf[n * GSTR + 8] = 0.f;
  }
  if (tid < RPB) {
    int r = blockIdx.x * RPB + tid;
    sdir[tid * 3 + 0] = rdir[r * 3 + 0];
    sdir[tid * 3 + 1] = rdir[r * 3 + 1];
    sdir[tid * 3 + 2] = rdir[r * 3 + 2];
  }
  __syncthreads();

  // ---------------- phase A: pair evaluation via V_WMMA_F32_16X16X4_F32 ----
  {
    const int m = lane & 15;       // A-matrix row (ray in block)
    const int h = lane >> 4;       // K half: VGPR0=K(2h), VGPR1=K(2h+1)
    const float d0 = sdir[m * 3 + 0];
    const float d1 = sdir[m * 3 + 1];
    const float d2 = sdir[m * 3 + 2];
    v2f aDot, aD1, aD2;
    aDot.x = h ? d2        : d0;        aDot.y = h ? 0.f       : d1;
    aD1.x  = h ? d2 * d2   : d0 * d0;   aD1.y  = h ? d0 * d1   : d1 * d1;
    aD2.x  = h ? 0.f       : d0 * d2;   aD2.y  = h ? 0.f       : d1 * d2;

    #pragma unroll
    for (int s = 0; s < 4; ++s) {
      const int n0  = (wv * 4 + s) * 16;
      const int col = n0 + m;
      const int k0  = 2 * h;
      v2f bDot, bD1, bD2;
      bDot.x = Bm[(0 + k0) * BSTR + col];
      bDot.y = Bm[(1 + k0) * BSTR + col];
      bD1.x  = Bm[(4 + k0) * BSTR + col];
      bD1.y  = Bm[(5 + k0) * BSTR + col];
      bD2.x  = Bm[(8 + k0) * BSTR + col];
      bD2.y  = Bm[(9 + k0) * BSTR + col];
      v8f z = {};
      v8f dotT = __builtin_amdgcn_wmma_f32_16x16x4_f32(
          false, aDot, false, bDot, (short)0, z, false, false);
      v8f ddT = __builtin_amdgcn_wmma_f32_16x16x4_f32(
          false, aD1, false, bD1, (short)0, z, false, false);
      ddT = __builtin_amdgcn_wmma_f32_16x16x4_f32(       // chain C accumulator
          false, aD2, false, bD2, (short)0, ddT, false, false);

      const float oo = gf[col * GSTR + 6];
      const float op = gf[col * GSTR + 7];
      #pragma unroll
      for (int v = 0; v < 8; ++v) {
        const int row = v + 8 * h;                 // D-matrix M index
        float dt = dotT[v], dd = ddT[v];
        float t  = -dt * __builtin_amdgcn_rcpf(dd);   // v_rcp_f32
        float dist2 = oo + dt * t;                    // oo - dt*dt/dd
        // alpha = min(opa * exp(-0.5*dist2), 0.999) via raw v_exp_f32 (exp2)
        float al = fminf(op * __builtin_amdgcn_exp2f(NEG_HALF_LOG2E * dist2),
                         0.999f);
        bool  val = (t > 0.f) && (al > 0.01f);
        float2 pr;
        pr.x = val ? t : __builtin_inff();
        // la = log2(1 - alpha); 1-al in [1e-3, 1] -> raw v_log_f32 is safe
        pr.y = val ? __builtin_amdgcn_logf(1.f - al) : 0.f;
        *(float2*)(tl + row * TS2 + 2 * col) = pr;    // ds_store_b64
      }
    }
  }
  __syncthreads();

  // ---------------- phase B: order-independent compositing (wave = ray) ----
  {
    const int r = blockIdx.x * RPB + wv;
    const float* tlr = tl + wv * TS2;
    unsigned long long ki[32];
    float La[32];
    #pragma unroll
    for (int k = 0; k < 32; ++k) {
      int i = lane + 32 * k;
      float2 p = *(const float2*)(tlr + 2 * i);       // ds_load_b64
      // non-negative floats: bit pattern order == value order; low 10 bits
      // carry the gaussian index for the stable (argsort) tie-break.
      ki[k] = ((unsigned long long)__float_as_uint(p.x) << 10) | (unsigned)i;
      La[k] = 0.f;
    }
    #pragma unroll 2
    for (int j = 0; j < NG; ++j) {
      float2 p = *(const float2*)(tlr + 2 * j);       // broadcast ds_load_b64
      unsigned long long kj =
          ((unsigned long long)__float_as_uint(p.x) << 10) | (unsigned)j;
      float lj = p.y;
      #pragma unroll
      for (int k = 0; k < 32; ++k) {
        La[k] += (kj < ki[k]) ? lj : 0.f;             // 1x v_cmp_lt_u64
      }
    }
    float im0 = 0.f, im1 = 0.f, im2 = 0.f;
    float na0 = 0.f, na1 = 0.f, na2 = 0.f;
    float dep = 0.f, asum = 0.f;
    #pragma unroll
    for (int k = 0; k < 32; ++k) {
      int i = lane + 32 * k;
      float2 p  = *(const float2*)(tlr + 2 * i);
      float lai = p.y;
      float T   = __builtin_amdgcn_exp2f(La[k]);   // transmittance before i
      float ai  = 1.f - __builtin_amdgcn_exp2f(lai); // alpha_i (0 if invalid)
      float wgt = (La[k] > LOG2_TMIN) ? ai * T : 0.f;  // early-stop gate
      float tv  = (lai < 0.f) ? p.x : 0.f;         // avoid 0*inf
      im0 += wgt * gf[i * GSTR + 0];
      im1 += wgt * gf[i * GSTR + 1];
      im2 += wgt * gf[i * GSTR + 2];
      na0 += wgt * gf[i * GSTR + 3];
      na1 += wgt * gf[i * GSTR + 4];
      na2 += wgt * gf[i * GSTR + 5];
      dep  += wgt * tv;
      asum += wgt;
    }
    im0 = wave_sum(im0); im1 = wave_sum(im1); im2 = wave_sum(im2);
    na0 = wave_sum(na0); na1 = wave_sum(na1); na2 = wave_sum(na2);
    dep = wave_sum(dep); asum = wave_sum(asum);
    if (lane == 0) {
      float Tf = 1.f - asum;
      float nn = rsqrtf(na0 * na0 + na1 * na1 + na2 * na2 + 1e-12f);
      out[r * 8 + 0] = im0 + Tf * bg[r * 3 + 0];
      out[r * 8 + 1] = im1 + Tf * bg[r * 3 + 1];
      out[r * 8 + 2] = im2 + Tf * bg[r * 3 + 2];
      out[r * 8 + 3] = asum;
      out[r * 8 + 4] = dep;
      out[r * 8 + 5] = na0 * nn;
      out[r * 8 + 6] = na1 * nn;
      out[r * 8 + 7] = na2 * nn;
    }
  }
}

extern "C" void kernel_launch(void* const* d_in, const int* in_sizes, int n_in,
                              void* d_out, int out_size, void* d_ws, size_t ws_size,
                              hipStream_t stream) {
  (void)in_sizes; (void)n_in; (void)out_size; (void)d_ws; (void)ws_size;
  const float* xyz  = (const float*)d_in[0];
  const float* rot  = (const float*)d_in[1];
  const float* sca  = (const float*)d_in[2];
  const float* opa  = (const float*)d_in[3];
  const float* rgb  = (const float*)d_in[4];
  const float* nrm  = (const float*)d_in[5];
  const float* bg   = (const float*)d_in[6];
  const float* rorg = (const float*)d_in[7];
  const float* rdir = (const float*)d_in[8];
  float* out = (float*)d_out;

  const size_t smem = (size_t)SM_TOT * sizeof(float);   // 218624 B (<320KB/WG)
  (void)hipFuncSetAttribute((const void*)gs_raytrace_kernel,
                            hipFuncAttributeMaxDynamicSharedMemorySize,
                            (int)smem);
  dim3 grid(16384 / RPB);   // 1024 blocks
  dim3 block(512);          // 16 wave32 waves
  gs_raytrace_kernel<<<grid, block, smem, stream>>>(
      xyz, rot, sca, opa, rgb, nrm, bg, rorg, rdir, out);
}